// TokenClassificationWithCRF_28965259444232
// MI455X (gfx1250) — compile-verified
//
#include <hip/hip_runtime.h>
#include <cstdint>

typedef __attribute__((ext_vector_type(2))) float v2f;
typedef __attribute__((ext_vector_type(8))) float v8f;

#define BB 32
#define SS 2048
#define HH 768
#define LL 9
#define NTOK (BB * SS)          // 65536 tokens
#define WSTRIDE 770             // padded LDS stride for W (768%64==0 is bank-pathological)
#define NCHUNK 64
#define TCH (SS / NCHUNK)       // 32 steps per chunk
#define NEGINF (-1e30f)

// ---------------------------------------------------------------------------
// Kernel 1: emissions = hidden_states @ W^T + b   via V_WMMA_F32_16X16X4_F32
// Each wave: 4 M-tiles (64 tokens) x 16 N (L=9 padded), K=768 in steps of 4.
// A-frag layout (16x4 f32): lane<16 -> M=lane, K={0,1}; lane>=16 -> M=lane-16, K={2,3}
// B-frag layout (4x16 f32): lane<16 -> N=lane, K={0,1}; lane>=16 -> N=lane-16, K={2,3}
// C layout: vgpr i: lanes 0-15 -> (M=i, N=lane); lanes 16-31 -> (M=8+i, N=lane-16)
// ---------------------------------------------------------------------------
__global__ __launch_bounds__(128)
void emis_gemm(const float* __restrict__ hs, const float* __restrict__ W,
               const float* __restrict__ bias, float* __restrict__ emis) {
  __shared__ float Wl[16 * WSTRIDE];
  const int tid = threadIdx.x;

  // Stage W (9x768, zero-padded to 16 rows) into LDS once per block.
  for (int idx = tid; idx < 16 * WSTRIDE; idx += 128) {
    int n = idx / WSTRIDE;
    int k = idx - n * WSTRIDE;
    float v = 0.0f;
    if (n < LL && k < HH) v = W[n * HH + k];
    Wl[idx] = v;
  }
  __syncthreads();

  const int wave  = (blockIdx.x * 128 + tid) >> 5;  // global wave id, 1024 total
  const int lane  = tid & 31;
  const int row   = lane & 15;
  const int khalf = lane >> 4;                      // 0: K={0,1}, 1: K={2,3}
  const long base = (long)wave * 64;                // 4 tiles of 16 tokens

  v8f acc0 = {}, acc1 = {}, acc2 = {}, acc3 = {};
  const float* h0 = hs + (base + 0 * 16 + row) * HH;
  const float* h1 = hs + (base + 1 * 16 + row) * HH;
  const float* h2 = hs + (base + 2 * 16 + row) * HH;
  const float* h3 = hs + (base + 3 * 16 + row) * HH;
  const float* wl = &Wl[row * WSTRIDE];

  for (int k0 = 0; k0 < HH; k0 += 4) {
    const int kk = k0 + 2 * khalf;
    v2f bf = *(const v2f*)(wl + kk);   // ds_load_b64, reused by 4 WMMAs
    v2f a0 = *(const v2f*)(h0 + kk);   // global_load_b64
    v2f a1 = *(const v2f*)(h1 + kk);
    v2f a2 = *(const v2f*)(h2 + kk);
    v2f a3 = *(const v2f*)(h3 + kk);
    acc0 = __builtin_amdgcn_wmma_f32_16x16x4_f32(false, a0, false, bf, (short)0, acc0, false, false);
    acc1 = __builtin_amdgcn_wmma_f32_16x16x4_f32(false, a1, false, bf, (short)0, acc1, false, false);
    acc2 = __builtin_amdgcn_wmma_f32_16x16x4_f32(false, a2, false, bf, (short)0, acc2, false, false);
    acc3 = __builtin_amdgcn_wmma_f32_16x16x4_f32(false, a3, false, bf, (short)0, acc3, false, false);
  }

  const int n = row;                  // output label index
  if (n < LL) {
    const float bn = bias[n];
#pragma unroll
    for (int i = 0; i < 8; ++i) {
      const int m = i + 8 * khalf;
      emis[(base + 0 * 16 + m) * LL + n] = acc0[i] + bn;
      emis[(base + 1 * 16 + m) * LL + n] = acc1[i] + bn;
      emis[(base + 2 * 16 + m) * LL + n] = acc2[i] + bn;
      emis[(base + 3 * 16 + m) * LL + n] = acc3[i] + bn;
    }
  }
}

// ---------------------------------------------------------------------------
// Kernel 2: per-chunk log-semiring transfer matrix (9x9) + gold-path partials.
// Block = (chunk c, batch b), 96 threads: tid<81 -> (s=tid/9, j=tid%9).
// Recursion: A'[s][j] = logsumexp_i(A[s][i] + T[i][j]) + em[t][j]; masked step
// leaves A unchanged (== multiply by log-identity).
// ---------------------------------------------------------------------------
__global__ __launch_bounds__(96)
void crf_chunk(const float* __restrict__ emis, const long long* __restrict__ labels,
               const float* __restrict__ trans, float* __restrict__ matws,
               float* __restrict__ scorews) {
  __shared__ float Ab[2][81];
  __shared__ float Tsh[81];
  __shared__ float red[32];

  const int tid = threadIdx.x;
  const int b = blockIdx.y, c = blockIdx.x;
  const int s = tid / 9, j = tid - s * 9;  // valid when tid < 81

  if (tid < 81) {
    Tsh[tid] = trans[tid];
    Ab[0][tid] = (s == j) ? 0.0f : NEGINF;  // log-semiring identity
  }
  __syncthreads();

  int tstart = c * TCH;
  if (tstart < 1) tstart = 1;              // step 0 is the init, chunk 0 starts at t=1
  const int tend = (c + 1) * TCH;
  const long long* lab = labels + (long)b * SS;
  const float* emb = emis + (long)b * SS * LL;

  int p = 0;
  for (int t = tstart; t < tend; ++t) {
    const bool mk = (lab[t] != -100LL);
    if (tid < 81) {
      float nv;
      if (mk) {
        const float e = emb[t * LL + j];
        float v[9];
        float m = NEGINF;
#pragma unroll
        for (int i = 0; i < 9; ++i) {
          v[i] = Ab[p][s * 9 + i] + Tsh[i * 9 + j];
          m = fmaxf(m, v[i]);
        }
        float sum = 0.0f;
#pragma unroll
        for (int i = 0; i < 9; ++i) sum += __expf(v[i] - m);
        nv = m + __logf(sum) + e;
      } else {
        nv = Ab[p][s * 9 + j];
      }
      Ab[1 - p][s * 9 + j] = nv;           // write to the other buffer: 1 barrier/step
    }
    __syncthreads();
    p ^= 1;
  }
  if (tid < 81) matws[((long)(b * NCHUNK) + c) * 81 + tid] = Ab[p][tid];

  // Gold-path score partial for this chunk (fixed-order reduction, deterministic).
  if (tid < 32) {
    float contrib = 0.0f;
    const int t = tstart + tid;
    if (t < tend) {
      const long long lt = lab[t], ltp = lab[t - 1];
      const int lc = (lt == -100LL) ? 0 : (int)lt;
      const int lp = (ltp == -100LL) ? 0 : (int)ltp;
      if (lt != -100LL) contrib = Tsh[lp * 9 + lc] + emb[t * LL + lc];
    }
    red[tid] = contrib;
  }
  __syncthreads();
  if (tid == 0) {
    float sum = 0.0f;
    for (int i = 0; i < 32; ++i) sum += red[i];
    scorews[b * NCHUNK + c] = sum;
  }
}

// ---------------------------------------------------------------------------
// Kernel 3: per-batch fold of 64 chunk matrices -> denom; boundary score terms;
// llh[b] = score - denom. One wave per batch.
// ---------------------------------------------------------------------------
__global__ __launch_bounds__(32)
void crf_combine(const float* __restrict__ emis, const long long* __restrict__ labels,
                 const float* __restrict__ startt, const float* __restrict__ endt,
                 const float* __restrict__ matws, const float* __restrict__ scorews,
                 float* __restrict__ llh) {
  __shared__ float al[9];
  const int b = blockIdx.x;
  const int j = threadIdx.x;
  const float* emb = emis + (long)b * SS * LL;
  const long long* lab = labels + (long)b * SS;

  if (j < 9) al[j] = startt[j] + emb[j];   // alpha_0
  __syncthreads();

  for (int c = 0; c < NCHUNK; ++c) {
    float nv = 0.0f;
    if (j < 9) {
      const float* M = matws + ((long)(b * NCHUNK) + c) * 81;
      float v[9];
      float m = NEGINF;
#pragma unroll
      for (int i = 0; i < 9; ++i) {
        v[i] = al[i] + M[i * 9 + j];
        m = fmaxf(m, v[i]);
      }
      float sum = 0.0f;
#pragma unroll
      for (int i = 0; i < 9; ++i) sum += __expf(v[i] - m);
      nv = m + __logf(sum);
    }
    __syncthreads();
    if (j < 9) al[j] = nv;
    __syncthreads();
  }

  // last valid tag position: strided scan + wave32 max-reduce
  int tmax = 0;
  for (int t = j; t < SS; t += 32)
    if (lab[t] != -100LL) tmax = t;
#pragma unroll
  for (int off = 16; off > 0; off >>= 1) {
    int o = __shfl_xor(tmax, off, 32);
    tmax = (o > tmax) ? o : tmax;
  }

  if (j == 0) {
    float v[9];
    float m = NEGINF;
    for (int i = 0; i < 9; ++i) { v[i] = al[i] + endt[i]; m = fmaxf(m, v[i]); }
    float sum = 0.0f;
    for (int i = 0; i < 9; ++i) sum += __expf(v[i] - m);
    const float denom = m + __logf(sum);

    float sc = 0.0f;
    for (int c = 0; c < NCHUNK; ++c) sc += scorews[b * NCHUNK + c];
    const long long l0 = lab[0];
    const int t0 = (l0 == -100LL) ? 0 : (int)l0;
    const long long llast = lab[tmax];
    const int tl = (llast == -100LL) ? 0 : (int)llast;
    sc += startt[t0] + emb[t0] + endt[tl];

    llh[b] = sc - denom;
  }
}

// ---------------------------------------------------------------------------
// Kernel 4: loss = -mean(llh)
// ---------------------------------------------------------------------------
__global__ __launch_bounds__(32)
void crf_loss(const float* __restrict__ llh, float* __restrict__ out) {
  if (threadIdx.x == 0) {
    float s = 0.0f;
    for (int i = 0; i < BB; ++i) s += llh[i];
    out[0] = -s / (float)BB;
  }
}

// ---------------------------------------------------------------------------
extern "C" void kernel_launch(void* const* d_in, const int* in_sizes, int n_in,
                              void* d_out, int out_size, void* d_ws, size_t ws_size,
                              hipStream_t stream) {
  const float*     hs     = (const float*)d_in[0];      // [B,S,H]
  const float*     W      = (const float*)d_in[1];      // [L,H]
  const float*     bias   = (const float*)d_in[2];      // [L]
  const float*     startt = (const float*)d_in[3];      // [L]
  const float*     endt   = (const float*)d_in[4];      // [L]
  const float*     trans  = (const float*)d_in[5];      // [L,L]
  const long long* labels = (const long long*)d_in[6];  // [B,S] int64
  // d_in[7] attention_mask: unused by the reference loss

  float* out  = (float*)d_out;
  float* emis = out + 1;                                // output: [loss, emissions...]

  float* matws   = (float*)d_ws;                        // B*NCHUNK*81 floats
  float* scorews = matws + (long)BB * NCHUNK * 81;      // B*NCHUNK floats
  float* llh     = scorews + BB * NCHUNK;               // B floats

  // 65536 tokens / 64 tokens-per-wave = 1024 waves; 4 waves/block -> 256 blocks
  emis_gemm<<<256, 128, 0, stream>>>(hs, W, bias, emis);
  crf_chunk<<<dim3(NCHUNK, BB), 96, 0, stream>>>(emis, labels, trans, matws, scorews);
  crf_combine<<<BB, 32, 0, stream>>>(emis, labels, startt, endt, matws, scorews, llh);
  crf_loss<<<1, 32, 0, stream>>>(llh, out);
}